// ModelActor_86784109183564
// MI455X (gfx1250) — compile-verified
//
#include <hip/hip_runtime.h>
#include <hip/hip_bf16.h>

typedef __attribute__((ext_vector_type(2))) float v2f;
typedef __attribute__((ext_vector_type(8))) float v8f;

#define Bb   16
#define Nn   5000
#define Ff   64
#define Hh   32
#define Ee   160000
#define Aa   16
#define NH   (Nn * Hh)      // 160000
#define HID  512
#define BNH  (Bb * Nn * Hh) // 2,560,000
#define KCHUNK 500          // 160000 / 320, multiple of 4

// ---------------------------------------------------------------------------
// Utility kernels
// ---------------------------------------------------------------------------
__global__ void fill_f32(float* __restrict__ p, float v, int n) {
  int i = blockIdx.x * blockDim.x + threadIdx.x;
  if (i < n) p[i] = v;
}

// deg[dst[e]] += 1  (deg pre-filled with 1.0 for the self loop)
__global__ void deg_accum(const int* __restrict__ dst, float* __restrict__ deg, int e) {
  int i = blockIdx.x * blockDim.x + threadIdx.x;
  if (i < e) atomicAdd(&deg[dst[i]], 1.0f);
}

// in-place deg -> deg^{-1/2}
__global__ void deg_to_dinv(float* __restrict__ deg, int n) {
  int i = blockIdx.x * blockDim.x + threadIdx.x;
  if (i < n) {
    float d = deg[i];
    deg[i] = (d > 0.0f) ? rsqrtf(d) : 0.0f;
  }
}

// y[i] = relu(y[i] + bias[i % hmod])   (in place)
__global__ void bias_relu(float* __restrict__ y, const float* __restrict__ bias,
                          int n, int hmod) {
  int i = blockIdx.x * blockDim.x + threadIdx.x;
  if (i < n) {
    float v = y[i] + bias[i % hmod];
    y[i] = v > 0.0f ? v : 0.0f;
  }
}

// ---------------------------------------------------------------------------
// GCN feature transform:  Y[M x 32] = X[M x K] @ W[K x 32]
// One wave per 16x16 output tile, fp32 WMMA 16x16x4, K in {64, 32}.
// A-frag per lane: row = lane&15, k = kk + 2*(lane>>4) + {0,1}
// B-frag per lane: col = lane&15, k = kk + 2*(lane>>4) + {0,1}
// D element v: (m = v + 8*(lane>>4), n = lane&15)
// ---------------------------------------------------------------------------
__global__ void gemm_xw_wmma(const float* __restrict__ X, const float* __restrict__ W,
                             float* __restrict__ Y, int K) {
  const int tm   = blockIdx.x;          // row tile (M/16)
  const int tn   = blockIdx.y;          // col tile (0..1)
  const int lane = threadIdx.x;
  const int half = lane >> 4;
  const int l15  = lane & 15;
  const int row  = tm * 16 + l15;
  const int col  = tn * 16 + l15;

  v8f acc = {};
  for (int kk = 0; kk < K; kk += 4) {
    const int ka = kk + half * 2;
    v2f a, b;
    a.x = X[(size_t)row * K + ka];
    a.y = X[(size_t)row * K + ka + 1];
    b.x = W[(size_t)ka * Hh + col];
    b.y = W[(size_t)(ka + 1) * Hh + col];
    acc = __builtin_amdgcn_wmma_f32_16x16x4_f32(false, a, false, b,
                                                (short)0, acc, false, false);
  }
#pragma unroll
  for (int v = 0; v < 8; ++v) {
    const int m = tm * 16 + v + 8 * half;
    Y[(size_t)m * Hh + tn * 16 + l15] = acc[v];
  }
}

// ---------------------------------------------------------------------------
// Edge scatter: out[:, dst, :] += xw[:, src, :] * dinv[src]*dinv[dst]
// One 128-thread block per edge; 512 (b,h) elements per edge.
// Working set (xw + out ~ 20 MB) is L2-resident on the 192 MB L2.
// ---------------------------------------------------------------------------
__global__ void scatter_edges(const int* __restrict__ ei, const float* __restrict__ dinv,
                              const float* __restrict__ xw, float* __restrict__ out) {
  const int e = blockIdx.x;             // 0 .. E+N-1
  int s, d;
  if (e < Ee) { s = ei[e]; d = ei[Ee + e]; }
  else        { s = d = e - Ee; }       // self loop
  const float norm = dinv[s] * dinv[d];

  for (int i = threadIdx.x; i < Bb * Hh; i += blockDim.x) {
    const int b = i >> 5;               // batch
    const int h = i & 31;               // feature
    const float val = xw[((size_t)b * Nn + s) * Hh + h] * norm;
    atomicAdd(&out[((size_t)b * Nn + d) * Hh + h], val);
  }
}

// ---------------------------------------------------------------------------
// MLP1 partial GEMM: hid[16 x 512] += flat[16 x 160000] @ Wp1[160000 x 512]
// K split into 320 chunks x 32 column tiles = 10240 waves to keep the
// 23.3 TB/s HBM pipe full while streaming the 327.7 MB Wp1 exactly once.
// Wp1 is larger than the 192 MB L2 and has zero reuse -> non-temporal loads,
// so the L2 keeps `flat` (10.2 MB, read by all 32 column tiles) resident.
// ---------------------------------------------------------------------------
__global__ void mlp1_wmma(const float* __restrict__ flat, const float* __restrict__ Wp1,
                          float* __restrict__ hid) {
  const int tn   = blockIdx.x;          // column tile 0..31
  const int kc   = blockIdx.y;          // K chunk 0..319
  const int lane = threadIdx.x;
  const int half = lane >> 4;
  const int l15  = lane & 15;
  const int col  = tn * 16 + l15;
  const int k0   = kc * KCHUNK;

  v8f acc = {};
  for (int kk = k0; kk < k0 + KCHUNK; kk += 4) {
    const int ka = kk + half * 2;
    v2f a, b;
    a.x = flat[(size_t)l15 * NH + ka];
    a.y = flat[(size_t)l15 * NH + ka + 1];
    b.x = __builtin_nontemporal_load(&Wp1[(size_t)ka * HID + col]);
    b.y = __builtin_nontemporal_load(&Wp1[(size_t)(ka + 1) * HID + col]);
    acc = __builtin_amdgcn_wmma_f32_16x16x4_f32(false, a, false, b,
                                                (short)0, acc, false, false);
  }
#pragma unroll
  for (int v = 0; v < 8; ++v) {
    const int m = v + 8 * half;
    atomicAdd(&hid[(size_t)m * HID + col], acc[v]);
  }
}

// ---------------------------------------------------------------------------
// MLP2: out[16 x 16] = hid[16 x 512] @ Wp2[512 x 16] + bp2   (single wave)
// ---------------------------------------------------------------------------
__global__ void mlp2_wmma(const float* __restrict__ hid, const float* __restrict__ Wp2,
                          const float* __restrict__ bp2, float* __restrict__ out) {
  const int lane = threadIdx.x;
  const int half = lane >> 4;
  const int l15  = lane & 15;

  v8f acc = {};
  for (int kk = 0; kk < HID; kk += 4) {
    const int ka = kk + half * 2;
    v2f a, b;
    a.x = hid[(size_t)l15 * HID + ka];
    a.y = hid[(size_t)l15 * HID + ka + 1];
    b.x = Wp2[(size_t)ka * Aa + l15];
    b.y = Wp2[(size_t)(ka + 1) * Aa + l15];
    acc = __builtin_amdgcn_wmma_f32_16x16x4_f32(false, a, false, b,
                                                (short)0, acc, false, false);
  }
#pragma unroll
  for (int v = 0; v < 8; ++v) {
    const int m = v + 8 * half;
    out[(size_t)m * Aa + l15] = acc[v] + bp2[l15];
  }
}

// ---------------------------------------------------------------------------
// Orchestration
// ---------------------------------------------------------------------------
extern "C" void kernel_launch(void* const* d_in, const int* in_sizes, int n_in,
                              void* d_out, int out_size, void* d_ws, size_t ws_size,
                              hipStream_t stream) {
  const float* features = (const float*)d_in[0];
  const int*   ei       = (const int*)d_in[1];   // int64 in ref, but JAX x64 off -> int32
  const float* W1       = (const float*)d_in[2];
  const float* b1       = (const float*)d_in[3];
  const float* W2       = (const float*)d_in[4];
  const float* b2       = (const float*)d_in[5];
  const float* Wp1      = (const float*)d_in[6];
  const float* bp1      = (const float*)d_in[7];
  const float* Wp2      = (const float*)d_in[8];
  const float* bp2      = (const float*)d_in[9];
  float* out = (float*)d_out;

  // workspace layout (floats)
  float* dinv = (float*)d_ws;           // [N]       (holds deg, then deg^-1/2)
  float* bufA = dinv + 5024;            // [B*N*H]   xw scratch
  float* bufB = bufA + BNH;             // [B*N*H]   aggregation / layer output
  float* hid  = bufB + BNH;             // [16*512]

  const int TB = 256;
  const int ET = Ee + Nn;               // edges incl. self loops

  // --- normalization coefficients ---
  fill_f32<<<(Nn + TB - 1) / TB, TB, 0, stream>>>(dinv, 1.0f, Nn);   // self loop
  deg_accum<<<(Ee + TB - 1) / TB, TB, 0, stream>>>(ei + Ee, dinv, Ee);
  deg_to_dinv<<<(Nn + TB - 1) / TB, TB, 0, stream>>>(dinv, Nn);

  // --- GCN layer 1 ---
  gemm_xw_wmma<<<dim3(Bb * Nn / 16, Hh / 16), 32, 0, stream>>>(features, W1, bufA, Ff);
  fill_f32<<<(BNH + TB - 1) / TB, TB, 0, stream>>>(bufB, 0.0f, BNH);
  scatter_edges<<<ET, 128, 0, stream>>>(ei, dinv, bufA, bufB);
  bias_relu<<<(BNH + TB - 1) / TB, TB, 0, stream>>>(bufB, b1, BNH, Hh);

  // --- GCN layer 2 ---
  gemm_xw_wmma<<<dim3(Bb * Nn / 16, Hh / 16), 32, 0, stream>>>(bufB, W2, bufA, Hh);
  fill_f32<<<(BNH + TB - 1) / TB, TB, 0, stream>>>(bufB, 0.0f, BNH);
  scatter_edges<<<ET, 128, 0, stream>>>(ei, dinv, bufA, bufB);
  bias_relu<<<(BNH + TB - 1) / TB, TB, 0, stream>>>(bufB, b2, BNH, Hh);

  // --- policy MLP ---
  fill_f32<<<(Bb * HID + TB - 1) / TB, TB, 0, stream>>>(hid, 0.0f, Bb * HID);
  mlp1_wmma<<<dim3(HID / 16, NH / KCHUNK), 32, 0, stream>>>(bufB, Wp1, hid);
  bias_relu<<<(Bb * HID + TB - 1) / TB, TB, 0, stream>>>(hid, bp1, Bb * HID, HID);
  mlp2_wmma<<<1, 32, 0, stream>>>(hid, Wp2, bp2, out);
}